// MambaClassifier_19653770347251
// MI455X (gfx1250) — compile-verified
//
#include <hip/hip_runtime.h>
#include <hip/hip_bf16.h>
#include <math.h>

// ---------------------------------------------------------------------------
// Vision-Mamba classifier for MI455X (gfx1250, wave32).
// Dense GEMMs use V_WMMA_F32_16X16X4_F32; large GEMMs register-block 32x32
// per wave (4 WMMAs per 4 fragment loads -> 8 FLOP/B, 2x the 16x16 tile).
// ---------------------------------------------------------------------------

typedef __attribute__((ext_vector_type(2))) float v2f;
typedef __attribute__((ext_vector_type(8))) float v8f;

#define ACT_NONE     0
#define ACT_SILU     1
#define ACT_SOFTPLUS 2

__device__ __forceinline__ float sigmoidf_(float x) {
    return 1.0f / (1.0f + __expf(-x));
}

__device__ __forceinline__ float apply_act(float val, int act) {
    if (act == ACT_SILU)          return val * sigmoidf_(val);
    if (act == ACT_SOFTPLUS)      return (val > 20.f) ? val : log1pf(__expf(val));
    return val;
}

// ---- 16x16-tile GEMM (generic fallback, used for N=48 x_proj) -------------
// C[m,n] = act( sum_k A[m,k] * W[n,k] + bias[n] (+ residual[m,n]) )
__global__ void gemm_wmma16_kernel(const float* __restrict__ A, int lda,
                                   const float* __restrict__ W,
                                   const float* __restrict__ bias,
                                   float* __restrict__ C, int ldc,
                                   const float* __restrict__ residual,
                                   int M, int N, int K, int act)
{
    const int tilesN = N >> 4;
    const int tiles  = (M >> 4) * tilesN;
    const int wid    = (int)((blockIdx.x * blockDim.x + threadIdx.x) >> 5);
    if (wid >= tiles) return;                 // whole-wave exit: EXEC stays all-1s
    const int lane = threadIdx.x & 31;
    const int half = lane >> 4;               // 0: K,K+1 ; 1: K+2,K+3
    const int lr   = lane & 15;
    const int tm = wid / tilesN, tn = wid % tilesN;
    const int m0 = tm << 4, n0 = tn << 4;

    const float* arow = A + (size_t)(m0 + lr) * lda + half * 2;
    const float* wrow = W + (size_t)(n0 + lr) * K   + half * 2;

    v8f acc = {0.f, 0.f, 0.f, 0.f, 0.f, 0.f, 0.f, 0.f};
    #pragma unroll 4
    for (int k = 0; k < K; k += 4) {
        v2f av = *(const v2f*)(arow + k);
        v2f bv = *(const v2f*)(wrow + k);
        acc = __builtin_amdgcn_wmma_f32_16x16x4_f32(
            false, av, false, bv, (short)0, acc, false, false);
    }

    const int col = n0 + lr;
    const float bv = bias ? bias[col] : 0.0f;
    #pragma unroll
    for (int v = 0; v < 8; ++v) {
        int row = m0 + v + half * 8;
        float val = acc[v] + bv;
        if (residual) val += residual[(size_t)row * ldc + col];
        C[(size_t)row * ldc + col] = apply_act(val, act);
    }
}

// ---- 32x32-tile GEMM (register-blocked; M%32==0, N%32==0) -----------------
// 2 A-fragments x 2 B-fragments per K-step -> 4 WMMAs per 4 b64 loads.
__global__ void gemm_wmma32_kernel(const float* __restrict__ A, int lda,
                                   const float* __restrict__ W,
                                   const float* __restrict__ bias,
                                   float* __restrict__ C, int ldc,
                                   const float* __restrict__ residual,
                                   int M, int N, int K, int act)
{
    const int tilesN = N >> 5;
    const int tiles  = (M >> 5) * tilesN;
    const int wid    = (int)((blockIdx.x * blockDim.x + threadIdx.x) >> 5);
    if (wid >= tiles) return;                 // whole-wave exit
    const int lane = threadIdx.x & 31;
    const int half = lane >> 4;
    const int lr   = lane & 15;
    const int tm = wid / tilesN, tn = wid % tilesN;
    const int m0 = tm << 5, n0 = tn << 5;

    const float* arow0 = A + (size_t)(m0 + lr) * lda + half * 2;
    const float* arow1 = arow0 + (size_t)16 * lda;
    const float* wrow0 = W + (size_t)(n0 + lr) * K + half * 2;
    const float* wrow1 = wrow0 + (size_t)16 * K;

    v8f acc00 = {0.f,0.f,0.f,0.f,0.f,0.f,0.f,0.f};
    v8f acc01 = acc00, acc10 = acc00, acc11 = acc00;

    #pragma unroll 2
    for (int k = 0; k < K; k += 4) {
        v2f a0 = *(const v2f*)(arow0 + k);
        v2f a1 = *(const v2f*)(arow1 + k);
        v2f b0 = *(const v2f*)(wrow0 + k);
        v2f b1 = *(const v2f*)(wrow1 + k);
        acc00 = __builtin_amdgcn_wmma_f32_16x16x4_f32(false, a0, false, b0, (short)0, acc00, false, false);
        acc01 = __builtin_amdgcn_wmma_f32_16x16x4_f32(false, a0, false, b1, (short)0, acc01, false, false);
        acc10 = __builtin_amdgcn_wmma_f32_16x16x4_f32(false, a1, false, b0, (short)0, acc10, false, false);
        acc11 = __builtin_amdgcn_wmma_f32_16x16x4_f32(false, a1, false, b1, (short)0, acc11, false, false);
    }

    const int c0 = n0 + lr, c1 = n0 + 16 + lr;
    const float bv0 = bias ? bias[c0] : 0.0f;
    const float bv1 = bias ? bias[c1] : 0.0f;
    #pragma unroll
    for (int v = 0; v < 8; ++v) {
        int r0 = m0 + v + half * 8;
        int r1 = r0 + 16;
        float v00 = acc00[v] + bv0, v01 = acc01[v] + bv1;
        float v10 = acc10[v] + bv0, v11 = acc11[v] + bv1;
        if (residual) {
            v00 += residual[(size_t)r0 * ldc + c0];
            v01 += residual[(size_t)r0 * ldc + c1];
            v10 += residual[(size_t)r1 * ldc + c0];
            v11 += residual[(size_t)r1 * ldc + c1];
        }
        C[(size_t)r0 * ldc + c0] = apply_act(v00, act);
        C[(size_t)r0 * ldc + c1] = apply_act(v01, act);
        C[(size_t)r1 * ldc + c0] = apply_act(v10, act);
        C[(size_t)r1 * ldc + c1] = apply_act(v11, act);
    }
}

// im2col for the 8x8/stride-8 patch embed: out[m, c*64 + i*8 + j]
__global__ void im2col_kernel(const float* __restrict__ x, float* __restrict__ col, int total)
{
    int idx = blockIdx.x * blockDim.x + threadIdx.x;
    if (idx >= total) return;
    int m  = idx / 192, c3 = idx % 192;
    int b  = m >> 10;
    int pr = m & 1023;
    int ph = pr >> 5, pw = pr & 31;
    int c  = c3 >> 6, r2 = c3 & 63;
    int i  = r2 >> 3, j  = r2 & 7;
    col[idx] = x[(((size_t)(b * 3 + c)) * 256 + (ph * 8 + i)) * 256 + (pw * 8 + j)];
}

// LayerNorm over D=256, one wave32 per token, 8 elements/lane.
__global__ void layernorm_kernel(const float* __restrict__ x, const float* __restrict__ g,
                                 const float* __restrict__ bb, float* __restrict__ o, int T)
{
    int wid = (int)((blockIdx.x * blockDim.x + threadIdx.x) >> 5);
    if (wid >= T) return;
    int lane = threadIdx.x & 31;
    const float* row = x + (size_t)wid * 256;
    float v[8]; float s = 0.f;
    #pragma unroll
    for (int j = 0; j < 8; ++j) { v[j] = row[lane + j * 32]; s += v[j]; }
    #pragma unroll
    for (int off = 16; off > 0; off >>= 1) s += __shfl_xor(s, off, 32);
    float mu = s * (1.f / 256.f);
    float vs = 0.f;
    #pragma unroll
    for (int j = 0; j < 8; ++j) { float d = v[j] - mu; vs += d * d; }
    #pragma unroll
    for (int off = 16; off > 0; off >>= 1) vs += __shfl_xor(vs, off, 32);
    float inv = rsqrtf(vs * (1.f / 256.f) + 1e-5f);
    #pragma unroll
    for (int j = 0; j < 8; ++j) {
        int cc = lane + j * 32;
        o[(size_t)wid * 256 + cc] = (v[j] - mu) * inv * g[cc] + bb[cc];
    }
}

// Causal depthwise conv (width 4) over L on the xb half of xz, then SiLU.
__global__ void dwconv_silu_kernel(const float* __restrict__ xz,
                                   const float* __restrict__ cw,
                                   const float* __restrict__ cb,
                                   float* __restrict__ out)
{
    int idx = blockIdx.x * blockDim.x + threadIdx.x;  // 4096*512
    if (idx >= 4096 * 512) return;
    int t = idx >> 9, d = idx & 511;
    int b = t >> 10, l = t & 1023;
    float w0 = cw[d * 4 + 0], w1 = cw[d * 4 + 1], w2 = cw[d * 4 + 2], w3 = cw[d * 4 + 3];
    float acc = cb[d];
    const float* base = xz + (size_t)(b * 1024) * 1024 + d;   // xb = cols 0..511 of xz
    if (l >= 3) acc += base[(size_t)(l - 3) * 1024] * w0;
    if (l >= 2) acc += base[(size_t)(l - 2) * 1024] * w1;
    if (l >= 1) acc += base[(size_t)(l - 1) * 1024] * w2;
    acc += base[(size_t)l * 1024] * w3;
    out[idx] = acc * sigmoidf_(acc);
}

// Fused selective scan + D*x skip + SiLU(z) gate.
// One thread per (b,d) channel; h[16] lives in registers; serial over L=1024.
__global__ void scan_kernel(const float* __restrict__ dtv, const float* __restrict__ xbc,
                            const float* __restrict__ dbc, const float* __restrict__ xz,
                            const float* __restrict__ A_log, const float* __restrict__ D_p,
                            float* __restrict__ yo)
{
    int gid = blockIdx.x * blockDim.x + threadIdx.x;  // 2048
    if (gid >= 2048) return;
    int b = gid >> 9, d = gid & 511;
    float Aneg[16];
    #pragma unroll
    for (int s = 0; s < 16; ++s) Aneg[s] = -__expf(A_log[d * 16 + s]);
    float h[16];
    #pragma unroll
    for (int s = 0; s < 16; ++s) h[s] = 0.f;
    float Dp = D_p[d];
    for (int l = 0; l < 1024; ++l) {
        int t = b * 1024 + l;
        float dt = dtv[(size_t)t * 512 + d];
        float xv = xbc[(size_t)t * 512 + d];
        float zz = xz[(size_t)t * 1024 + 512 + d];         // z = cols 512..1023 of xz
        const float* bc = dbc + (size_t)t * 48;            // [dtr(16) | B(16) | C(16)]
        float dtx = dt * xv;
        float y = 0.f;
        #pragma unroll
        for (int s = 0; s < 16; ++s) {
            float dA = __expf(dt * Aneg[s]);
            h[s] = dA * h[s] + dtx * bc[16 + s];
            y += h[s] * bc[32 + s];
        }
        y += Dp * xv;
        yo[(size_t)t * 512 + d] = y * (zz * sigmoidf_(zz));
    }
}

// Mean over L=1024 tokens; one block per batch, one thread per model dim.
__global__ void meanpool_kernel(const float* __restrict__ x, float* __restrict__ o)
{
    int b = blockIdx.x, dm = threadIdx.x;  // 4 x 256
    float s = 0.f;
    for (int l = 0; l < 1024; ++l) s += x[((size_t)(b * 1024 + l)) * 256 + dm];
    o[b * 256 + dm] = s * (1.f / 1024.f);
}

// Tiny classifier head: 4x1000, K=256.
__global__ void classifier_kernel(const float* __restrict__ p, const float* __restrict__ w,
                                  const float* __restrict__ bb, float* __restrict__ out)
{
    int idx = blockIdx.x * blockDim.x + threadIdx.x;
    if (idx >= 4000) return;
    int b = idx / 1000, n = idx % 1000;
    const float* pr = p + b * 256;
    const float* wr = w + (size_t)n * 256;
    float s = bb[n];
    #pragma unroll 4
    for (int k = 0; k < 256; ++k) s += pr[k] * wr[k];
    out[idx] = s;
}

extern "C" void kernel_launch(void* const* d_in, const int* in_sizes, int n_in,
                              void* d_out, int out_size, void* d_ws, size_t ws_size,
                              hipStream_t stream)
{
    const float* x       = (const float*)d_in[0];
    const float* patch_w = (const float*)d_in[1];
    const float* patch_b = (const float*)d_in[2];
    const float* ln_g    = (const float*)d_in[3];
    const float* ln_b    = (const float*)d_in[4];
    const float* in_w    = (const float*)d_in[5];
    const float* conv_w  = (const float*)d_in[6];
    const float* conv_b  = (const float*)d_in[7];
    const float* xproj_w = (const float*)d_in[8];
    const float* dt_w    = (const float*)d_in[9];
    const float* dt_b    = (const float*)d_in[10];
    const float* A_log   = (const float*)d_in[11];
    const float* D_p     = (const float*)d_in[12];
    const float* out_w   = (const float*)d_in[13];
    const float* fn_g    = (const float*)d_in[14];
    const float* fn_b    = (const float*)d_in[15];
    const float* cls_w   = (const float*)d_in[16];
    const float* cls_b   = (const float*)d_in[17];
    (void)in_sizes; (void)n_in; (void)out_size; (void)ws_size;

    float* ws = (float*)d_ws;
    size_t off = 0;
    float* u      = ws + off; off += (size_t)4096 * 256;
    float* lnb    = ws + off; off += (size_t)4096 * 256;
    float* xz     = ws + off; off += (size_t)4096 * 1024;
    float* xbc    = ws + off; off += (size_t)4096 * 512;
    float* dbc    = ws + off; off += (size_t)4096 * 48;
    float* dtv    = ws + off; off += (size_t)4096 * 512;
    float* ybuf   = ws + off; off += (size_t)4096 * 512;
    float* pooled = ws + off; off += 1024;
    float* col    = xz;  // im2col only used before xz is first written

    auto gemm = [&](const float* A, int lda, const float* W, const float* bias,
                    float* C, int ldc, const float* res, int M, int N, int K, int act) {
        if ((M % 32 == 0) && (N % 32 == 0)) {
            int tiles  = (M / 32) * (N / 32);
            int blocks = (tiles + 7) / 8;  // 8 waves per 256-thread block
            gemm_wmma32_kernel<<<blocks, 256, 0, stream>>>(A, lda, W, bias, C, ldc, res, M, N, K, act);
        } else {
            int tiles  = (M / 16) * (N / 16);
            int blocks = (tiles + 7) / 8;
            gemm_wmma16_kernel<<<blocks, 256, 0, stream>>>(A, lda, W, bias, C, ldc, res, M, N, K, act);
        }
    };

    // Patch embedding: im2col + GEMM [4096x192]x[192x256] + bias
    im2col_kernel<<<(4096 * 192 + 255) / 256, 256, 0, stream>>>(x, col, 4096 * 192);
    gemm(col, 192, patch_w, patch_b, u, 256, nullptr, 4096, 256, 192, ACT_NONE);

    for (int i = 0; i < 4; ++i) {
        // LayerNorm(u)
        layernorm_kernel<<<512, 256, 0, stream>>>(u, ln_g + i * 256, ln_b + i * 256, lnb, 4096);
        // in_proj: xz[4096,1024]
        gemm(lnb, 256, in_w + (size_t)i * 1024 * 256, nullptr, xz, 1024, nullptr,
             4096, 1024, 256, ACT_NONE);
        // depthwise causal conv + SiLU -> xbc[4096,512]
        dwconv_silu_kernel<<<(4096 * 512) / 256, 256, 0, stream>>>(
            xz, conv_w + (size_t)i * 512 * 4, conv_b + i * 512, xbc);
        // x_proj: dbc[4096,48] = [dtr | B | C]  (N=48 -> 16x16 kernel)
        gemm(xbc, 512, xproj_w + (size_t)i * 48 * 512, nullptr, dbc, 48, nullptr,
             4096, 48, 512, ACT_NONE);
        // dt = softplus(dtr @ dt_w^T + dt_b) -> dtv[4096,512]
        gemm(dbc, 48, dt_w + (size_t)i * 512 * 16, dt_b + i * 512, dtv, 512, nullptr,
             4096, 512, 16, ACT_SOFTPLUS);
        // fused selective scan + skip + gate -> ybuf[4096,512]
        scan_kernel<<<8, 256, 0, stream>>>(dtv, xbc, dbc, xz,
                                           A_log + (size_t)i * 512 * 16, D_p + i * 512, ybuf);
        // out_proj + residual (in place on u)
        gemm(ybuf, 512, out_w + (size_t)i * 256 * 512, nullptr, u, 256, u,
             4096, 256, 512, ACT_NONE);
    }

    // Final LN -> mean pool -> classifier
    layernorm_kernel<<<512, 256, 0, stream>>>(u, fn_g, fn_b, lnb, 4096);
    meanpool_kernel<<<4, 256, 0, stream>>>(lnb, pooled);
    classifier_kernel<<<16, 256, 0, stream>>>(pooled, cls_w, cls_b, (float*)d_out);
}